// AnchorManager_29798483100025
// MI455X (gfx1250) — compile-verified
//
#include <hip/hip_runtime.h>
#include <hip/hip_bf16.h>

#define SSD_N 100
#define BLK   256
#define SSD_EPS 1e-6f

#if defined(__HIP_DEVICE_COMPILE__) && __has_builtin(__builtin_amdgcn_global_load_async_to_lds_b128) && __has_builtin(__builtin_amdgcn_global_load_async_to_lds_b32) && __has_builtin(__builtin_amdgcn_s_wait_asynccnt)
#define USE_ASYNC_LDS 1
#else
#define USE_ASYNC_LDS 0
#endif

#if USE_ASYNC_LDS
// Builtin signatures (from hipcc diagnostics):
//   global_load_async_to_lds_b128(v4i AS1*, v4i AS3*, Imm int, Imm int)
//   global_load_async_to_lds_b32 (int AS1*, int AS3*, Imm int, Imm int)
typedef int v4i_t __attribute__((__vector_size__(16)));
typedef __attribute__((address_space(1))) v4i_t* gp128_t;
typedef __attribute__((address_space(3))) v4i_t* lp128_t;
typedef __attribute__((address_space(1))) int*   gp32_t;
typedef __attribute__((address_space(3))) int*   lp32_t;
#endif

__device__ __forceinline__ unsigned long long umax64(unsigned long long a, unsigned long long b) {
    return a > b ? a : b;
}

__device__ __forceinline__ float iou_pair(float ax1, float ay1, float ax2, float ay2, float area_a,
                                          float gx1, float gy1, float gx2, float gy2, float area_g) {
    float ltx = fmaxf(ax1, gx1), lty = fmaxf(ay1, gy1);
    float rbx = fminf(ax2, gx2), rby = fminf(ay2, gy2);
    float w = fmaxf(rbx - ltx, 0.0f), h = fmaxf(rby - lty, 0.0f);
    float inter = w * h;
    return inter / (area_a + area_g - inter);
}

// Stage one batch's GT boxes (N rows of 16B) into LDS.
// Uses CDNA5 async global->LDS DMA (ASYNCcnt) when available.
__device__ __forceinline__ void stage_gt_boxes(const float* __restrict__ gsrc, float* s_gt, int tid) {
#if USE_ASYNC_LDS
    if (tid < SSD_N) {
        float* g = const_cast<float*>(gsrc) + tid * 4;
        __builtin_amdgcn_global_load_async_to_lds_b128(
            (gp128_t)g, (lp128_t)(s_gt + tid * 4), 0, 0);
    }
    __builtin_amdgcn_s_wait_asynccnt(0);
#else
    if (tid < SSD_N) {
        float4 v = ((const float4*)gsrc)[tid];
        ((float4*)s_gt)[tid] = v;
    }
#endif
}

// ---------------- Kernel 1: per-anchor max/argmax over GTs ----------------
__global__ void k_anchor_best(const float* __restrict__ gt_boxes,     // [B,N,4] xyxy
                              const float* __restrict__ anchors_xyxy, // [A,4]
                              unsigned long long* __restrict__ packed, // [B*A] {iou_bits, n}
                              int A) {
    __shared__ __align__(16) float s_gt[SSD_N * 4];
    __shared__ float s_area[SSD_N];
    const int b = blockIdx.y;
    const int tid = threadIdx.x;

    stage_gt_boxes(gt_boxes + (size_t)b * SSD_N * 4, s_gt, tid);
    __syncthreads();
    if (tid < SSD_N) {
        float gx1 = s_gt[tid * 4 + 0], gy1 = s_gt[tid * 4 + 1];
        float gx2 = s_gt[tid * 4 + 2], gy2 = s_gt[tid * 4 + 3];
        s_area[tid] = (gx2 - gx1) * (gy2 - gy1);
    }
    __syncthreads();

    const int a = blockIdx.x * BLK + tid;
    if (a >= A) return;

    const float4 av = ((const float4*)anchors_xyxy)[a];
    const float area_a = (av.z - av.x) * (av.w - av.y);

    float best = -1.0f;
    int bestn = 0;
    #pragma unroll 4
    for (int n = 0; n < SSD_N; ++n) {
        float iou = iou_pair(av.x, av.y, av.z, av.w, area_a,
                             s_gt[n * 4 + 0], s_gt[n * 4 + 1],
                             s_gt[n * 4 + 2], s_gt[n * 4 + 3], s_area[n]);
        if (iou > best) { best = iou; bestn = n; }  // strict > : first-index tiebreak
    }
    packed[(size_t)b * A + a] =
        ((unsigned long long)__float_as_uint(best) << 32) | (unsigned int)bestn;
}

// ---------------- Kernel 2: per-GT argmax over anchors + force match ----------------
__global__ void k_gt_best(const float* __restrict__ gt_boxes,
                          const float* __restrict__ anchors_xyxy,
                          unsigned long long* __restrict__ packed,
                          int A) {
    __shared__ unsigned long long red[BLK];
    const int n = blockIdx.x;
    const int b = blockIdx.y;
    const int tid = threadIdx.x;

    const float* g = gt_boxes + ((size_t)b * SSD_N + n) * 4;  // uniform -> scalar loads
    const float gx1 = g[0], gy1 = g[1], gx2 = g[2], gy2 = g[3];
    const float area_g = (gx2 - gx1) * (gy2 - gy1);

    unsigned long long key = 0;  // iou>=0 so any real candidate beats 0
    for (int a = tid; a < A; a += BLK) {
        if (a + BLK < A) __builtin_prefetch(anchors_xyxy + 4 * (a + BLK), 0, 0);
        const float4 av = ((const float4*)anchors_xyxy)[a];
        const float area_a = (av.z - av.x) * (av.w - av.y);
        float iou = iou_pair(av.x, av.y, av.z, av.w, area_a, gx1, gy1, gx2, gy2, area_g);
        // max-key: higher iou wins; on ties, lower anchor index wins (~a trick)
        unsigned long long k = ((unsigned long long)__float_as_uint(iou) << 32) |
                               (unsigned long long)(0xFFFFFFFFu - (unsigned int)a);
        key = umax64(key, k);
    }
    red[tid] = key;
    __syncthreads();
    #pragma unroll
    for (int s = BLK / 2; s > 0; s >>= 1) {
        if (tid < s) red[tid] = umax64(red[tid], red[tid + s]);
        __syncthreads();
    }
    if (tid == 0) {
        unsigned int a_star = 0xFFFFFFFFu - (unsigned int)(red[0] & 0xFFFFFFFFull);
        // single 64-bit store keeps {value,index} consistent under duplicate a_star races
        packed[(size_t)b * A + a_star] =
            ((unsigned long long)__float_as_uint(2.0f) << 32) | (unsigned int)n;
    }
}

// ---------------- Kernel 3: gather + encode + emit outputs ----------------
__global__ void k_encode(const int* __restrict__ gt_labels,      // [B,N]
                         const float* __restrict__ gt_boxes,     // [B,N,4] xyxy
                         const float* __restrict__ anchors_c,    // [A,4] cxcywh
                         const unsigned long long* __restrict__ packed,
                         float* __restrict__ out, int A, int B) {
    __shared__ __align__(16) float s_gt[SSD_N * 4];
    __shared__ int s_lab[SSD_N];
    const int b = blockIdx.y;
    const int tid = threadIdx.x;

    stage_gt_boxes(gt_boxes + (size_t)b * SSD_N * 4, s_gt, tid);
#if USE_ASYNC_LDS
    if (tid < SSD_N) {
        int* gl = const_cast<int*>(gt_labels) + (size_t)b * SSD_N + tid;
        __builtin_amdgcn_global_load_async_to_lds_b32(
            (gp32_t)gl, (lp32_t)(s_lab + tid), 0, 0);
    }
    __builtin_amdgcn_s_wait_asynccnt(0);
#else
    if (tid < SSD_N) s_lab[tid] = gt_labels[(size_t)b * SSD_N + tid];
#endif
    __syncthreads();

    const int a = blockIdx.x * BLK + tid;
    if (a >= A) return;

    const unsigned long long p = packed[(size_t)b * A + a];
    const float val = __uint_as_float((unsigned int)(p >> 32));
    const int   n   = (int)(unsigned int)(p & 0xFFFFFFFFull);
    const bool  pos = val > 0.5f;

    const float x1 = s_gt[n * 4 + 0], y1 = s_gt[n * 4 + 1];
    const float x2 = s_gt[n * 4 + 2], y2 = s_gt[n * 4 + 3];
    const float mcx = (x1 + x2) * 0.5f, mcy = (y1 + y2) * 0.5f;
    const float mw = x2 - x1, mh = y2 - y1;

    const float4 anc = ((const float4*)anchors_c)[a];
    float4 e;
    e.x = (mcx - anc.x) / anc.z;
    e.y = (mcy - anc.y) / anc.w;
    e.z = logf((mw + SSD_EPS) / (anc.z + SSD_EPS));
    e.w = logf((mh + SSD_EPS) / (anc.w + SSD_EPS));

    const size_t BA = (size_t)B * A;
    const size_t i = (size_t)b * A + a;
    out[i] = pos ? (float)s_lab[n] : 0.0f;   // encoded_labels (background=0)
    ((float4*)(out + BA))[i] = e;            // encoded [B,A,4]
    out[BA * 5 + i] = pos ? 1.0f : 0.0f;     // pos_mask
}

extern "C" void kernel_launch(void* const* d_in, const int* in_sizes, int n_in,
                              void* d_out, int out_size, void* d_ws, size_t ws_size,
                              hipStream_t stream) {
    const int*   gt_labels    = (const int*)d_in[0];
    const float* gt_boxes     = (const float*)d_in[1];
    const float* anchors_c    = (const float*)d_in[2];
    const float* anchors_xyxy = (const float*)d_in[3];

    const int B = in_sizes[0] / SSD_N;   // gt_labels flat = B*N
    const int A = in_sizes[2] / 4;       // anchors flat = A*4

    unsigned long long* packed = (unsigned long long*)d_ws;  // B*A u64, fully rewritten each call

    dim3 blk(BLK);
    dim3 gA((A + BLK - 1) / BLK, B);
    dim3 gN(SSD_N, B);

    k_anchor_best<<<gA, blk, 0, stream>>>(gt_boxes, anchors_xyxy, packed, A);
    k_gt_best   <<<gN, blk, 0, stream>>>(gt_boxes, anchors_xyxy, packed, A);
    k_encode    <<<gA, blk, 0, stream>>>(gt_labels, gt_boxes, anchors_c, packed,
                                         (float*)d_out, A, B);
}